// Com2Net_33973191311415
// MI455X (gfx1250) — compile-verified
//
#include <hip/hip_runtime.h>

// ---------------------------------------------------------------------------
// Com2Net on gfx1250 — runs-batched recurrent MLP, V_WMMA_F32_16X16X4_F32.
//
//  layer1: D[hidden(16pad) x runs(16)] = W1[16x4] * inp^T[4x16] + b1  (WMMA,
//          bias fused into the C accumulator)
//  tanh  : v_tanh_f32 on the 8 accumulator VGPRs (padded rows == tanh(0)=0)
//  layer2: 3x10 dot in VALU; rows 8,9 cross the half-wave via ds_bpermute
//
//  Recurrence shortcuts (the chain is the whole cost of this problem):
//   * comm[2i]   (forward channel)  = previous agent's o2  -> register
//                forward via shfl_xor(o2,16), no LDS round trip.
//   * comm[2i+3] (backward channel) = previous-timestep data -> LDS load
//                issued one iteration ahead (never waits in the chain).
//   * x inputs  -> global load issued one iteration ahead.
//   * fully branchless inner loop: operand halves picked with v_cndmask,
//     lanes 16-31 store to dummy LDS rows (no exec save/restore).
//  Only the ODD comm slots need storage: slot[r][i] == comm[2i+1].
// ---------------------------------------------------------------------------

typedef float v2f __attribute__((ext_vector_type(2)));
typedef float v8f __attribute__((ext_vector_type(8)));

#define R_TOT          128
#define T_STEPS        1024
#define N_AG           128
#define RUNS_PER_BLOCK 16
#define SLOT_STRIDE    132   // >=130 slots; 132*4B = 16B multiple, bank stagger 4

__device__ __forceinline__ float fast_tanh(float x) {
#if __has_builtin(__builtin_amdgcn_tanhf)
  return __builtin_amdgcn_tanhf(x);                    // v_tanh_f32 (TRANS)
#else
  float e = __builtin_exp2f(x * 2.885390081777927f);   // v_exp_f32 fallback
  return 1.0f - 2.0f / (e + 1.0f);
#endif
}

__global__ __launch_bounds__(32)
void com2net_wmma_kernel(const float* __restrict__ runs,
                         const float* __restrict__ W1,
                         const float* __restrict__ b1,
                         const float* __restrict__ W2,
                         const float* __restrict__ b2,
                         float*       __restrict__ out)
{
  // Rows 0-15: real state for runs runBase..runBase+15. Rows 16-31: dummy
  // sinks so upper-lane stores need no exec masking.
  __shared__ __align__(16) float cslot [32 * SLOT_STRIDE];  // odd comm slots
  __shared__ __align__(16) float outbuf[32 * SLOT_STRIDE];  // per-t controls

  const int lane    = threadIdx.x;      // 0..31 (wave32)
  const int up      = lane >> 4;        // 0: lanes 0-15, 1: lanes 16-31
  const int l15     = lane & 15;
  const int runBase = blockIdx.x * RUNS_PER_BLOCK;

  // comm starts at zero every call (slots 128,129 stay zero forever)
  for (int k = lane; k < 32 * SLOT_STRIDE; k += 32) cslot[k] = 0.0f;
  // single wave per block: DScnt ordering, no barriers anywhere

  // ---- A operand: W1 zero-padded to 16x4 (M = hidden unit, K = feature) ----
  // A-layout (16x4 f32): lane = M; VGPR0 = K0 (lanes 0-15)/K2 (lanes 16-31),
  //                      VGPR1 = K1/K3.
  v2f amat; amat.x = 0.0f; amat.y = 0.0f;
  {
    const int h  = l15;
    const int k0 = up * 2;
    if (h < 10) { amat.x = W1[h * 4 + k0]; amat.y = W1[h * 4 + k0 + 1]; }
  }

  // ---- C operand: b1 broadcast along columns: C[m][n] = b1[m] ----
  v8f bias1;
#pragma unroll
  for (int v = 0; v < 8; ++v) {
    const int m = v + 8 * up;
    bias1[v] = (m < 10) ? b1[m] : 0.0f;
  }

  // ---- layer-2 weights (uniform -> SGPRs) ----
  float w2r[30];
#pragma unroll
  for (int k = 0; k < 30; ++k) w2r[k] = W2[k];
  const float b20 = b2[0], b21 = b2[1], b22 = b2[2];

  const float* xrow_base = runs + (size_t)(runBase + l15) * T_STEPS * (N_AG * 2);
  const float* cread     = &cslot [l15  * SLOT_STRIDE];  // real row (run l15)
  float*       cwrite    = &cslot [lane * SLOT_STRIDE];  // dummy row if up
  float*       obwrite   = &outbuf[lane * SLOT_STRIDE];  // dummy row if up

  for (int t = 0; t < T_STEPS; ++t) {
    const float* xrow = xrow_base + (size_t)t * (N_AG * 2);

    // L2 prefetch of next timestep's input row (1 KB/run, 2 lanes/run).
    if (t + 1 < T_STEPS) {
      const char* nxt = (const char*)(xrow + N_AG * 2);
#pragma unroll
      for (int j = 0; j < 4; ++j)
        __builtin_prefetch(nxt + (size_t)(up * 4 + j) * 128, 0, 0);
    }

    // ---- pipeline prologue for i = 0 ----
    v2f   xv_cur = *(const v2f*)xrow;   // x(i=0)
    float cb_cur = cread[1];            // comm[3]  (backward channel, prev t)
    float co     = 0.0f;                // comm[0] is never written -> 0

    for (int i = 0; i < N_AG; ++i) {
      // ---- issue next iteration's independent loads (hide their latency) --
      const int ip = (i + 1 < N_AG) ? i + 1 : i;          // clamp (no OOB)
      const v2f   xv_next = *(const v2f*)(xrow + ip * 2); // LOADcnt
      const float cb_next = cread[ip + 1];                // DScnt (prev-t data)

      // ---- B operand: inp^T (4 x 16 runs), branchless select ----
      // K0,K1 rows (lanes 0-15): x0,x1.  K2,K3 rows (lanes 16-31): commA,commB.
      v2f bmat;
      bmat.x = up ? co     : xv_cur.x;   // v_cndmask
      bmat.y = up ? cb_cur : xv_cur.y;

      // D[hidden x runs] = W1 * inp^T + b1
      v8f d = __builtin_amdgcn_wmma_f32_16x16x4_f32(
          false, amat, false, bmat, (short)0, bias1, false, false);

      float dt[8];
#pragma unroll
      for (int v = 0; v < 8; ++v) dt[v] = fast_tanh(d[v]);

      // rows 8,9 of my run live in lane^16, VGPRs 0,1
      const float h8 = __shfl_xor(dt[0], 16, 32);
      const float h9 = __shfl_xor(dt[1], 16, 32);

      // ---- o2 first: it is the recurrence (2-way split FMA tree) ----
      float e0 = fmaf(w2r[20], dt[0], b22);
      float e1 =      w2r[21] * dt[1];
      e0 = fmaf(w2r[22], dt[2], e0);  e1 = fmaf(w2r[23], dt[3], e1);
      e0 = fmaf(w2r[24], dt[4], e0);  e1 = fmaf(w2r[25], dt[5], e1);
      e0 = fmaf(w2r[26], dt[6], e0);  e1 = fmaf(w2r[27], dt[7], e1);
      e0 = fmaf(w2r[28], h8,    e0);  e1 = fmaf(w2r[29], h9,    e1);
      const float o2 = e0 + e1;
      co = __shfl_xor(o2, 16, 32);    // forward channel for iteration i+1

      // ---- o0, o1 (off the recurrence) ----
      float a0 = fmaf(w2r[0], dt[0], b20), a1 = w2r[1] * dt[1];
      a0 = fmaf(w2r[2], dt[2], a0);  a1 = fmaf(w2r[3], dt[3], a1);
      a0 = fmaf(w2r[4], dt[4], a0);  a1 = fmaf(w2r[5], dt[5], a1);
      a0 = fmaf(w2r[6], dt[6], a0);  a1 = fmaf(w2r[7], dt[7], a1);
      a0 = fmaf(w2r[8], h8,   a0);   a1 = fmaf(w2r[9], h9,   a1);
      const float o0 = a0 + a1;

      float c0 = fmaf(w2r[10], dt[0], b21), c1 = w2r[11] * dt[1];
      c0 = fmaf(w2r[12], dt[2], c0); c1 = fmaf(w2r[13], dt[3], c1);
      c0 = fmaf(w2r[14], dt[4], c0); c1 = fmaf(w2r[15], dt[5], c1);
      c0 = fmaf(w2r[16], dt[6], c0); c1 = fmaf(w2r[17], dt[7], c1);
      c0 = fmaf(w2r[18], h8,   c0);  c1 = fmaf(w2r[19], h9,   c1);
      const float o1 = c0 + c1;

      // branchless stores: lanes 16-31 hit dummy rows
      obwrite[i] = o0;                // control -> staged row
      cwrite[i]  = o1;                // comm[2i+1], read next timestep

      xv_cur = xv_next;               // rotate pipeline
      cb_cur = cb_next;
    }

    // ---- coalesced flush of this timestep's controls ----
#pragma unroll 4
    for (int rr = 0; rr < RUNS_PER_BLOCK; ++rr) {
      const float4 vv = *(const float4*)&outbuf[rr * SLOT_STRIDE + lane * 4];
      *(float4*)(out + ((size_t)(runBase + rr) * T_STEPS + t) * N_AG + lane * 4) = vv;
    }
  }
}

extern "C" void kernel_launch(void* const* d_in, const int* in_sizes, int n_in,
                              void* d_out, int out_size, void* d_ws, size_t ws_size,
                              hipStream_t stream) {
  (void)in_sizes; (void)n_in; (void)out_size; (void)d_ws; (void)ws_size;
  const float* runs = (const float*)d_in[0];
  const float* W1   = (const float*)d_in[1];
  const float* b1   = (const float*)d_in[2];
  const float* W2   = (const float*)d_in[3];
  const float* b2   = (const float*)d_in[4];
  float* out        = (float*)d_out;

  dim3 grid(R_TOT / RUNS_PER_BLOCK);   // 8 independent 16-run chains
  dim3 block(32);                      // one wave32 per block
  com2net_wmma_kernel<<<grid, block, 0, stream>>>(runs, W1, b1, W2, b2, out);
}